// GIN_25202868093368
// MI455X (gfx1250) — compile-verified
//
#include <hip/hip_runtime.h>
#include <hip/hip_bf16.h>
#include <math.h>

typedef float v2f __attribute__((ext_vector_type(2)));
typedef float v8f __attribute__((ext_vector_type(8)));

#define DH 128

// ---------------- copy (agg init: agg = x, folds the self term) ----------------
__global__ __launch_bounds__(256)
void gin_copy_f4(const float4* __restrict__ src, float4* __restrict__ dst, int n4) {
  int i = blockIdx.x * blockDim.x + threadIdx.x;
  if (i < n4) dst[i] = src[i];
}

// ---------------- scatter-add: Agg[dst] += X[src], one wave per edge ----------------
__global__ __launch_bounds__(256)
void gin_scatter_add(const float* __restrict__ X, const int* __restrict__ src,
                     const int* __restrict__ dst, float* __restrict__ Agg, int E) {
  int e = blockIdx.x * 8 + (threadIdx.x >> 5);
  if (e >= E) return;
  int lane = threadIdx.x & 31;
  int s = src[e];
  int d = dst[e];
  const float* xs = X + (size_t)s * DH;
  float*       ag = Agg + (size_t)d * DH;
#pragma unroll
  for (int j = 0; j < 4; ++j)
    atomicAdd(&ag[lane + 32 * j], xs[lane + 32 * j]);
}

// ---------------- C = relu(A @ W + b), fp32 WMMA 16x16x4 ----------------
// 256 threads = 8 waves. Block computes 64 rows x 128 cols.
// Wave w owns columns [16w, 16w+16): preloads its 128x16 W slab as 32 v2f
// B-fragments in registers; A band staged in LDS via async global->LDS copies.
__global__ __launch_bounds__(256)
void gin_gemm_bias_relu(const float* __restrict__ A, const float* __restrict__ W,
                        const float* __restrict__ bias, float* __restrict__ C, int M) {
  __shared__ float As[64][132];   // +4 pad: conflict-free column reads (132 % 64 == 4)

  const int tid   = threadIdx.x;
  const int wave  = tid >> 5;
  const int lane  = tid & 31;
  const int nl    = lane & 15;    // N (and M) index within fragment
  const int khalf = lane >> 4;    // selects K pair {0,1} vs {2,3}
  const int row0  = blockIdx.x * 64;
  const int n0    = wave * 16;

  // Stage the 64x128 A band straight into LDS with async copies (no VGPR bounce).
  // Rows >= M are clamped to row M-1: row i of C depends only on row i of A, and
  // those output rows are never stored, so the duplicated data is harmless.
  for (int i = tid; i < 64 * 32; i += 256) {
    int r = i >> 5, c4 = i & 31;
    int grow = row0 + r;
    if (grow >= M) grow = M - 1;
    unsigned lds_dst = (unsigned)(size_t)&As[r][c4 * 4];      // addr[31:0] == LDS offset
    const float* gsrc = A + (size_t)grow * DH + c4 * 4;
    asm volatile("global_load_async_to_lds_b128 %0, %1, off"
                 :: "v"(lds_dst), "v"(gsrc)
                 : "memory");
  }

  // Preload B fragments: b = { W[k + 2*khalf][n], W[k + 2*khalf + 1][n] } per ISA layout.
  v2f Wn[32];
#pragma unroll
  for (int ks = 0; ks < 32; ++ks) {
    int r = ks * 4 + 2 * khalf;
    Wn[ks].x = W[(size_t)r * DH + n0 + nl];
    Wn[ks].y = W[(size_t)(r + 1) * DH + n0 + nl];
  }
  float bcol = bias[n0 + nl];

  asm volatile("s_wait_asynccnt 0x0" ::: "memory");
  __syncthreads();

#pragma unroll
  for (int mt = 0; mt < 4; ++mt) {
    v8f acc = {};
    const float* arow = &As[mt * 16 + nl][0];   // A fragment: M = lane & 15
#pragma unroll
    for (int ks = 0; ks < 32; ++ks) {
      int kc = ks * 4 + 2 * khalf;
      v2f a;
      a.x = arow[kc];
      a.y = arow[kc + 1];
      // 8 args: (neg_a, A, neg_b, B, c_mod, C, reuse_a, reuse_b)
      acc = __builtin_amdgcn_wmma_f32_16x16x4_f32(false, a, false, Wn[ks],
                                                  (short)0, acc, false, false);
    }
    // D layout: VGPR r -> row r + 8*khalf, col = n0 + (lane & 15)
#pragma unroll
    for (int r = 0; r < 8; ++r) {
      int grow = row0 + mt * 16 + r + 8 * khalf;
      if (grow < M) {
        float v = acc[r] + bcol;
        C[(size_t)grow * DH + n0 + nl] = v > 0.f ? v : 0.f;
      }
    }
  }
}

// ---------------- out = sigmoid(H @ Wfc + bfc), one wave per node ----------------
__global__ __launch_bounds__(256)
void gin_fc_sigmoid(const float* __restrict__ H, const float* __restrict__ Wfc,
                    const float* __restrict__ bfc, float* __restrict__ out, int M) {
  int node = blockIdx.x * 8 + (threadIdx.x >> 5);
  int lane = threadIdx.x & 31;
  if (node >= M) return;
  const float* h = H + (size_t)node * DH;
  float s = 0.f;
#pragma unroll
  for (int j = 0; j < 4; ++j)
    s += h[lane * 4 + j] * Wfc[lane * 4 + j];
#pragma unroll
  for (int off = 16; off > 0; off >>= 1)
    s += __shfl_down(s, off, 32);
  if (lane == 0)
    out[node] = 1.f / (1.f + __expf(-(s + bfc[0])));
}

extern "C" void kernel_launch(void* const* d_in, const int* in_sizes, int n_in,
                              void* d_out, int out_size, void* d_ws, size_t ws_size,
                              hipStream_t stream) {
  (void)n_in; (void)out_size; (void)ws_size;

  const float* x   = (const float*)d_in[0];
  const int*   ei  = (const int*)d_in[1];    // (2, E) int32
  const float* W1a = (const float*)d_in[2];
  const float* b1a = (const float*)d_in[3];
  const float* W1b = (const float*)d_in[4];
  const float* b1b = (const float*)d_in[5];
  const float* W2a = (const float*)d_in[6];
  const float* b2a = (const float*)d_in[7];
  const float* W2b = (const float*)d_in[8];
  const float* b2b = (const float*)d_in[9];
  const float* Wfc = (const float*)d_in[10];
  const float* bfc = (const float*)d_in[11];
  float* out = (float*)d_out;

  const int M = in_sizes[0] / DH;   // 50000 nodes
  const int E = in_sizes[1] / 2;    // 800000 edges
  const int* src = ei;
  const int* dst = ei + E;

  float* bufA = (float*)d_ws;                      // 25.6 MB each
  float* bufB = bufA + (size_t)M * DH;
  float* bufC = bufB + (size_t)M * DH;

  const int n4       = M * (DH / 4);
  const dim3 blk(256);
  const int copyGrid = (n4 + 255) / 256;
  const int scatGrid = (E + 7) / 8;
  const int gemmGrid = (M + 63) / 64;
  const int fcGrid   = (M + 7) / 8;

  // ---- GIN conv 1 ----
  gin_copy_f4      <<<copyGrid, blk, 0, stream>>>((const float4*)x, (float4*)bufA, n4);
  gin_scatter_add  <<<scatGrid, blk, 0, stream>>>(x, src, dst, bufA, E);
  gin_gemm_bias_relu<<<gemmGrid, blk, 0, stream>>>(bufA, W1a, b1a, bufB, M);
  gin_gemm_bias_relu<<<gemmGrid, blk, 0, stream>>>(bufB, W1b, b1b, bufC, M);

  // ---- GIN conv 2 ----
  gin_copy_f4      <<<copyGrid, blk, 0, stream>>>((const float4*)bufC, (float4*)bufA, n4);
  gin_scatter_add  <<<scatGrid, blk, 0, stream>>>(bufC, src, dst, bufA, E);
  gin_gemm_bias_relu<<<gemmGrid, blk, 0, stream>>>(bufA, W2a, b2a, bufB, M);
  gin_gemm_bias_relu<<<gemmGrid, blk, 0, stream>>>(bufB, W2b, b2b, bufC, M);

  // ---- FC + sigmoid ----
  gin_fc_sigmoid   <<<fcGrid, blk, 0, stream>>>(bufC, Wfc, bfc, out, M);
}